// PointHead_10831907520977
// MI455X (gfx1250) — compile-verified
//
#include <hip/hip_runtime.h>
#include <math.h>

// ---------------- problem constants (from setup_inputs) ----------------
#define B_    8      // batch
#define NC    19     // coarse channels (mlp output rows too)
#define HC    48     // coarse H=W
#define P_    48     // points per batch (N)
#define NOG   144    // over-generation candidates (3*48)
#define NIMP  36     // importance points (0.75*48)
#define NCOV  12     // coverage points
#define CF    256    // fine channels
#define HF    192    // fine H=W
#define CT    275    // feat channels = 19+256
#define CTP   276    // K padded to multiple of 4 for wmma 16x16x4
#define MPAD  32     // M (=19) padded to two 16-row tiles

typedef __attribute__((ext_vector_type(2))) float v2f;
typedef __attribute__((ext_vector_type(8))) float v8f;

// ---------------------------------------------------------------------
// Kernel 0: zero-pad mlp_w [19,275] -> wpad [32,276] so the GEMM K-loop
// has no bounds checks (no EXEC save/restore in the hot loop).
// ---------------------------------------------------------------------
__global__ void pad_weights_kernel(const float* __restrict__ w,
                                   float* __restrict__ wpad) {
    int i = blockIdx.x * blockDim.x + threadIdx.x;   // 0 .. 32*276-1
    if (i >= MPAD * CTP) return;
    int r = i / CTP, c = i % CTP;
    wpad[i] = (r < NC && c < CT) ? w[(size_t)r * CT + c] : 0.f;
}

// ---------------------------------------------------------------------
// Kernel 1: per-batch point selection.
// uncertainty(p) = bilinear(top2map) - bilinear(top1map) at candidate p,
// then stable top-36 by rank counting (ties -> lower index, like lax.top_k).
// ---------------------------------------------------------------------
__device__ __forceinline__ void top2_pixel(const float* __restrict__ outp,
                                           int b, int yi, int xi,
                                           float& t1, float& t2) {
    const float* base = outp + (size_t)b * NC * HC * HC + (size_t)yi * HC + xi;
    float m1 = -3.4e38f, m2 = -3.4e38f;
#pragma unroll
    for (int c = 0; c < NC; ++c) {
        float v = base[(size_t)c * HC * HC];
        if (v > m1) { m2 = m1; m1 = v; }
        else if (v > m2) { m2 = v; }
    }
    t1 = m1; t2 = m2;
}

__global__ void select_points_kernel(const float* __restrict__ outp,
                                     const float* __restrict__ og,
                                     const float* __restrict__ cov,
                                     float* __restrict__ ws_points,
                                     float* __restrict__ out_points) {
    __shared__ float u[NOG];
    const int b = blockIdx.x;
    const int t = threadIdx.x;

    if (t < NOG) {
        float px = og[((size_t)b * NOG + t) * 2 + 0];
        float py = og[((size_t)b * NOG + t) * 2 + 1];
        float gx = px * (float)HC - 0.5f;
        float gy = py * (float)HC - 0.5f;
        float fx = floorf(gx), fy = floorf(gy);
        float wx = gx - fx, wy = gy - fy;
        int x0 = (int)fx, y0 = (int)fy;
        float s0 = 0.f, s1 = 0.f;
#pragma unroll
        for (int dy = 0; dy < 2; ++dy) {
#pragma unroll
            for (int dx = 0; dx < 2; ++dx) {
                int xi = x0 + dx, yi = y0 + dy;
                if (xi >= 0 && xi < HC && yi >= 0 && yi < HC) {
                    float w = (dx ? wx : 1.f - wx) * (dy ? wy : 1.f - wy);
                    float t1, t2;
                    top2_pixel(outp, b, yi, xi, t1, t2);
                    s0 += w * t1;
                    s1 += w * t2;
                }
            }
        }
        u[t] = s1 - s0;   // uncertainty = -(top1 - top2)
    }
    __syncthreads();

    if (t < NOG) {
        float ui = u[t];
        int rank = 0;
        for (int j = 0; j < NOG; ++j) {
            float uj = u[j];
            if (uj > ui || (uj == ui && j < t)) ++rank;
        }
        if (rank < NIMP) {
            float px = og[((size_t)b * NOG + t) * 2 + 0];
            float py = og[((size_t)b * NOG + t) * 2 + 1];
            int o = b * P_ * 2 + rank * 2;
            ws_points[o]     = px; ws_points[o + 1] = py;
            out_points[o]    = px; out_points[o + 1] = py;
        }
    } else if (t < NOG + NCOV) {
        int j = t - NOG;
        float px = cov[((size_t)b * NCOV + j) * 2 + 0];
        float py = cov[((size_t)b * NCOV + j) * 2 + 1];
        int o = b * P_ * 2 + (NIMP + j) * 2;
        ws_points[o]     = px; ws_points[o + 1] = py;
        out_points[o]    = px; out_points[o + 1] = py;
    }
}

// ---------------------------------------------------------------------
// Kernel 2: build feat in K-pair interleaved layout:
//   feat[b][pair = c>>1][p][c&1],  pair in [0,138), p in [0,48)
// so the WMMA B-fragment per-lane (K=kk, K=kk+1) pair at column n is one
// contiguous, 8B-aligned float2 -> single global_load_b64 in the GEMM.
// Channel 275 is the zero K-pad.
// ---------------------------------------------------------------------
__global__ void build_feat_kernel(const float* __restrict__ outp,
                                  const float* __restrict__ res2,
                                  const float* __restrict__ ws_points,
                                  float* __restrict__ feat) {
    const int bp = blockIdx.x;
    const int b = bp / P_, p = bp % P_;
    const float px = ws_points[((size_t)b * P_ + p) * 2 + 0];
    const float py = ws_points[((size_t)b * P_ + p) * 2 + 1];

    // coarse grid (48x48)
    float gxc = px * (float)HC - 0.5f, gyc = py * (float)HC - 0.5f;
    float fxc = floorf(gxc), fyc = floorf(gyc);
    float wxc = gxc - fxc, wyc = gyc - fyc;
    int x0c = (int)fxc, y0c = (int)fyc;

    // fine grid (192x192)
    float gxf = px * (float)HF - 0.5f, gyf = py * (float)HF - 0.5f;
    float fxf = floorf(gxf), fyf = floorf(gyf);
    float wxf = gxf - fxf, wyf = gyf - fyf;
    int x0f = (int)fxf, y0f = (int)fyf;

    float* fb = feat + (size_t)b * CTP * P_;

    for (int c = threadIdx.x; c < CTP; c += blockDim.x) {
        float val = 0.f;
        if (c < NC) {
            const float* base = outp + ((size_t)b * NC + c) * HC * HC;
#pragma unroll
            for (int dy = 0; dy < 2; ++dy)
#pragma unroll
                for (int dx = 0; dx < 2; ++dx) {
                    int xi = x0c + dx, yi = y0c + dy;
                    if (xi >= 0 && xi < HC && yi >= 0 && yi < HC) {
                        float w = (dx ? wxc : 1.f - wxc) * (dy ? wyc : 1.f - wyc);
                        val += w * base[(size_t)yi * HC + xi];
                    }
                }
        } else if (c < CT) {
            const float* base = res2 + ((size_t)b * CF + (c - NC)) * HF * HF;
#pragma unroll
            for (int dy = 0; dy < 2; ++dy)
#pragma unroll
                for (int dx = 0; dx < 2; ++dx) {
                    int xi = x0f + dx, yi = y0f + dy;
                    if (xi >= 0 && xi < HF && yi >= 0 && yi < HF) {
                        float w = (dx ? wxf : 1.f - wxf) * (dy ? wyf : 1.f - wyf);
                        val += w * base[(size_t)yi * HF + xi];
                    }
                }
        }
        // K-pair interleaved store
        fb[(size_t)(c >> 1) * (P_ * 2) + p * 2 + (c & 1)] = val;
    }
}

// ---------------------------------------------------------------------
// Kernel 3: rend[b] = mlp_w(19x275) @ feat[b](275x48) + bias via
// V_WMMA_F32_16X16X4_F32. One wave per 16x16 output tile.
// A 16x4 frag:  lanes 0-15 -> K=k,k+1 ; lanes 16-31 -> K=k+2,k+3 ; M=lane%16
// B 4x16 frag:  lanes 0-15 -> K=k,k+1 ; lanes 16-31 -> K=k+2,k+3 ; N=lane%16
// C 16x16 frag: vgpr r -> row r (lanes 0-15) / row 8+r (lanes 16-31)
// Both operand loads are unconditional 8B-aligned b64 loads (padded data).
// Bias is folded into the C-fragment initialization.
// ---------------------------------------------------------------------
__global__ void gemm_wmma_kernel(const float* __restrict__ wpad,
                                 const float* __restrict__ bias,
                                 const float* __restrict__ feat,
                                 float* __restrict__ rend) {
    const int tile = blockIdx.x;        // 0..5
    const int b    = blockIdx.y;        // 0..7
    const int mt = tile / 3, nt = tile % 3;
    const int lane = threadIdx.x;       // wave32, full EXEC
    const int half = lane >> 4;         // 0: K pair (k,k+1), 1: (k+2,k+3)
    const int lm   = lane & 15;

    v8f acc = {};
#pragma unroll
    for (int r = 0; r < 8; ++r) {
        int row = mt * 16 + r + half * 8;
        acc[r] = (row < NC) ? bias[row] : 0.f;
    }

    const int n    = nt * 16 + lm;      // output column / B-matrix N
    const int rowm = mt * 16 + lm;      // A-matrix row M (padded array: no check)

    // per-lane starting pointers; advance by 4 K per iteration
    const float* ap = wpad + (size_t)rowm * CTP + half * 2;
    const float* bp = feat + (size_t)b * CTP * P_ + (size_t)half * (P_ * 2) + n * 2;

    for (int k = 0; k < CTP; k += 4) {
        v2f av = *(const v2f*)ap;                 // A[rowm][kk], A[rowm][kk+1]
        v2f bv = *(const v2f*)bp;                 // feat[kk][n], feat[kk+1][n]
        // 8 args: (neg_a, A, neg_b, B, c_mod, C, reuse_a, reuse_b)
        acc = __builtin_amdgcn_wmma_f32_16x16x4_f32(
            false, av, false, bv, (short)0, acc, false, false);
        ap += 4;                // 4 K values per step along the weight row
        bp += 2 * (P_ * 2);     // 2 K-pairs per step in pair-interleaved feat
    }

#pragma unroll
    for (int r = 0; r < 8; ++r) {
        int row = mt * 16 + r + half * 8;
        if (row < NC)
            rend[((size_t)b * NC + row) * P_ + n] = acc[r];
    }
}

// ---------------------------------------------------------------------
extern "C" void kernel_launch(void* const* d_in, const int* in_sizes, int n_in,
                              void* d_out, int out_size, void* d_ws, size_t ws_size,
                              hipStream_t stream) {
    // inputs: 0:x (unused), 1:res2, 2:out, 3:over_generation, 4:coverage,
    //         5:mlp_w, 6:mlp_b
    const float* res2 = (const float*)d_in[1];
    const float* outp = (const float*)d_in[2];
    const float* og   = (const float*)d_in[3];
    const float* cov  = (const float*)d_in[4];
    const float* w    = (const float*)d_in[5];
    const float* bias = (const float*)d_in[6];

    float* rend       = (float*)d_out;                  // [8,19,48] = 7296 floats
    float* out_points = rend + (size_t)B_ * NC * P_;    // [8,48,2]  =  768 floats

    float* ws_points = (float*)d_ws;                    // B*48*2   =    768 floats
    float* feat      = ws_points + (size_t)B_ * P_ * 2; // B*276*48 = 105984 floats
    float* wpad      = feat + (size_t)B_ * CTP * P_;    // 32*276   =   8832 floats

    pad_weights_kernel<<<(MPAD * CTP + 255) / 256, 256, 0, stream>>>(w, wpad);
    select_points_kernel<<<B_, 160, 0, stream>>>(outp, og, cov, ws_points, out_points);
    build_feat_kernel<<<B_ * P_, 64, 0, stream>>>(outp, res2, ws_points, feat);
    gemm_wmma_kernel<<<dim3(6, B_), 32, 0, stream>>>(wpad, bias, feat, rend);
}